// GivensSSMCuda_17102559773347
// MI455X (gfx1250) — compile-verified
//
#include <hip/hip_runtime.h>
#include <math.h>

// Problem constants (match reference)
#define Bc    8
#define Tc    4096
#define Dc    512
#define Pc    256
#define BT    (Bc * Tc)        // 32768 rows
#define CHUNK 128
#define NCH   (Tc / CHUNK)     // 32 chunks per sequence
#define NTILES 48              // 768 output cols (256 angle + 512 inj) / 16
#define KQ2   64               // 512 K / 8 (two WMMAs per packed load)

typedef float v2f __attribute__((ext_vector_type(2)));
typedef float v4f __attribute__((ext_vector_type(4)));
typedef float v8f __attribute__((ext_vector_type(8)));

// ---------------------------------------------------------------------------
// Kernel 0: repack [W_angle | W_input] into per-lane WMMA B-fragment order.
// packW[ntile][kq2][lane] = float4 { B(k0+2h), B(k0+2h+1), B(k0+4+2h), B(k0+4+2h+1) }
// so the GEMM loop does one coalesced global_load_b128 per two WMMAs.
// 1.5 MB total — L2-resident, one-time cost.
// ---------------------------------------------------------------------------
__global__ __launch_bounds__(256) void packW_kernel(
    const float* __restrict__ Wang,  // [512, 256]
    const float* __restrict__ Winp,  // [512, 512]
    float* __restrict__ packW)       // [48*64*32] float4
{
    const int e = blockIdx.x * 256 + threadIdx.x;    // over NTILES*KQ2*32
    if (e >= NTILES * KQ2 * 32) return;
    const int lane  = e & 31;
    const int kq2   = (e >> 5) & (KQ2 - 1);
    const int ntile = e >> 11;                       // / (KQ2*32)

    const int  n0     = ntile * 16;
    const bool is_ang = (n0 < 256);
    const float* W    = is_ang ? Wang : Winp;
    const int  ldw    = is_ang ? 256 : 512;
    const int  col    = (is_ang ? n0 : (n0 - 256)) + (lane & 15);
    const int  khalf  = lane >> 4;
    const int  kb0    = kq2 * 8 + 2 * khalf;         // first WMMA of the pair
    const int  kb1    = kb0 + 4;                     // second WMMA of the pair

    float4 v;
    v.x = W[(kb0    ) * ldw + col];
    v.y = W[(kb0 + 1) * ldw + col];
    v.z = W[(kb1    ) * ldw + col];
    v.w = W[(kb1 + 1) * ldw + col];
    ((float4*)packW)[e] = v;
}

// ---------------------------------------------------------------------------
// Kernel 1: fused projection GEMM via V_WMMA_F32_16X16X4_F32.
// Computes [angles | inj] = x @ [W_angle | W_input] + [b_angle | b_input].
// One block owns a 16-row slab of x (staged once in LDS); its 8 waves sweep
// all 48 16-column tiles (6 per wave), so x is read from HBM exactly once.
// B fragments come from packW: one global_load_b128 feeds two WMMAs.
// inj columns are written directly into d_out (consumed in place by the scan).
// ---------------------------------------------------------------------------
__global__ __launch_bounds__(256) void proj_wmma_kernel(
    const float* __restrict__ x,      // [BT, 512]
    const float* __restrict__ packW,  // packed B fragments
    const float* __restrict__ bang,   // [256]
    const float* __restrict__ binp,   // [512]
    float* __restrict__ angles,       // [BT, 256]   (workspace)
    float* __restrict__ inj_out)      // [BT, 512]   (= d_out)
{
    __shared__ float xs[16 * Dc];     // 32 KB slab: rows m0..m0+15 of x
    const int m0 = blockIdx.x * 16;

    for (int i = threadIdx.x; i < 16 * Dc; i += 256)
        xs[i] = x[m0 * Dc + i];
    __syncthreads();

    const int lane  = threadIdx.x & 31;
    const int wv    = threadIdx.x >> 5;   // 0..7
    const int rowl  = lane & 15;          // A-matrix M within tile
    const int khalf = lane >> 4;          // 0: K={0,1}, 1: K={2,3}

    for (int nsub = 0; nsub < 6; ++nsub) {
        const int ntile = nsub * 8 + wv;        // 0..47
        const int n0    = ntile * 16;
        const bool is_ang = (n0 < 256);
        const int  coll   = (is_ang ? n0 : (n0 - 256)) + (lane & 15);

        const v4f* bp = (const v4f*)packW + ((size_t)ntile * KQ2 + 0) * 32 + lane;

        v8f acc = {0.f, 0.f, 0.f, 0.f, 0.f, 0.f, 0.f, 0.f};
        for (int kq2 = 0; kq2 < KQ2; ++kq2) {
            const int k0 = kq2 * 8;
            v2f af0, af1;
            af0.x = xs[rowl * Dc + k0 + 2 * khalf];
            af0.y = xs[rowl * Dc + k0 + 2 * khalf + 1];
            af1.x = xs[rowl * Dc + k0 + 4 + 2 * khalf];
            af1.y = xs[rowl * Dc + k0 + 4 + 2 * khalf + 1];

            const v4f bq = bp[(size_t)kq2 * 32];     // global_load_b128, coalesced
            v2f bf0 = {bq.x, bq.y};
            v2f bf1 = {bq.z, bq.w};

            acc = __builtin_amdgcn_wmma_f32_16x16x4_f32(
                false, af0, false, bf0, (short)0, acc, false, false);
            acc = __builtin_amdgcn_wmma_f32_16x16x4_f32(
                false, af1, false, bf1, (short)0, acc, false, false);
        }

        const float bias = (is_ang ? bang : binp)[coll];
        float*      dst  = is_ang ? angles : inj_out;
        const int   ldd  = is_ang ? 256 : 512;
        #pragma unroll
        for (int r = 0; r < 8; ++r) {
            const int grow = m0 + r + 8 * khalf;   // C/D layout: VGPR r -> M=r / M=r+8
            dst[grow * ldd + coll] = acc[r] + bias;
        }
    }
}

// ---------------------------------------------------------------------------
// Kernel 2: decay = sigmoid(x @ W_decay + b_decay), one thread per row.
// ---------------------------------------------------------------------------
__global__ __launch_bounds__(256) void decay_kernel(
    const float* __restrict__ x, const float* __restrict__ Wd,
    const float* __restrict__ bd, float* __restrict__ decays)
{
    const int row = blockIdx.x * 256 + threadIdx.x;
    if (row >= BT) return;
    const float4* xr = (const float4*)(x + (size_t)row * Dc);
    const float4* wr = (const float4*)Wd;
    float acc = 0.f;
    #pragma unroll 4
    for (int i = 0; i < Dc / 4; ++i) {
        const float4 a = xr[i], b = wr[i];
        acc += a.x * b.x + a.y * b.y + a.z * b.z + a.w * b.w;
    }
    acc += bd[0];
    decays[row] = 1.f / (1.f + __expf(-acc));
}

// ---------------------------------------------------------------------------
// Kernel 3: chunk-local scan with zero initial state.
// Block = one (batch, chunk); thread = one pair p. Writes local states into
// d_out in place (each inj element is read exactly once, then overwritten by
// the same thread). angles is overwritten with in-chunk prefix sums (cumA),
// cumD holds in-chunk decay prefix products, sumA the chunk angle totals.
// ---------------------------------------------------------------------------
__global__ __launch_bounds__(256) void scan_local_kernel(
    float* __restrict__ angles,        // [BT, 256]  in: a, out: cumA
    const float* __restrict__ decays,  // [BT]
    float* __restrict__ cumD,          // [BT]
    float* __restrict__ sumA,          // [B*NCH*P]
    float* __restrict__ out)           // [BT, 512]  in: inj, out: local state
{
    const int b  = blockIdx.x / NCH;
    const int ch = blockIdx.x % NCH;
    const int p  = threadIdx.x;
    const int t0 = ch * CHUNK;

    float h0 = 0.f, h1 = 0.f, cA = 0.f, cD = 1.f;
    for (int i = 0; i < CHUNK; ++i) {
        const int t    = t0 + i;
        const int rid  = b * Tc + t;
        const int aidx = rid * Pc + p;

        const float  a = angles[aidx];
        const float  d = decays[rid];
        const float2 u = *(const float2*)(out + (size_t)rid * Dc + 2 * p);

        if (i + 4 < CHUNK) {
            __builtin_prefetch(&angles[aidx + 4 * Pc], 0, 1);      // global_prefetch_b8
            __builtin_prefetch(out + (size_t)(rid + 4) * Dc + 2 * p, 0, 1);
        }

        float s, c;
        __sincosf(a, &s, &c);
        const float he = c * h0 - s * h1;
        const float ho = s * h0 + c * h1;
        h0 = d * he + u.x;
        h1 = d * ho + u.y;

        *(float2*)(out + (size_t)rid * Dc + 2 * p) = make_float2(h0, h1);

        cA += a;
        cD *= d;
        angles[aidx] = cA;
        if (p == 0) cumD[rid] = cD;
    }
    sumA[(b * NCH + ch) * Pc + p] = cA;
}

// ---------------------------------------------------------------------------
// Kernel 4: compose the 32 chunk aggregates per (b, p) to get each chunk's
// incoming state h_in.  h_in[c+1] = prodD_c * R(sumA_c) h_in[c] + local_last_c.
// ---------------------------------------------------------------------------
__global__ __launch_bounds__(256) void scan_carry_kernel(
    const float* __restrict__ sumA, const float* __restrict__ cumD,
    const float* __restrict__ out, float* __restrict__ hin)
{
    const int b = blockIdx.x;
    const int p = threadIdx.x;
    float h0 = 0.f, h1 = 0.f;
    for (int ch = 0; ch < NCH; ++ch) {
        *(float2*)(hin + ((size_t)(b * NCH + ch) * Pc + p) * 2) = make_float2(h0, h1);
        const int   tlast = ch * CHUNK + CHUNK - 1;
        const int   rid   = b * Tc + tlast;
        const float sA    = sumA[(b * NCH + ch) * Pc + p];
        const float pD    = cumD[rid];
        const float2 loc  = *(const float2*)(out + (size_t)rid * Dc + 2 * p);
        float s, c;
        __sincosf(sA, &s, &c);
        const float n0 = pD * (c * h0 - s * h1) + loc.x;
        const float n1 = pD * (s * h0 + c * h1) + loc.y;
        h0 = n0; h1 = n1;
    }
}

// ---------------------------------------------------------------------------
// Kernel 5: fully-parallel fixup: h_t += cumD_t * R(cumA_t) * h_in(chunk).
// ---------------------------------------------------------------------------
__global__ __launch_bounds__(256) void fixup_kernel(
    const float* __restrict__ angles,  // now cumA
    const float* __restrict__ cumD,
    const float* __restrict__ hin,
    float* __restrict__ out)
{
    const long long idx = (long long)blockIdx.x * 256 + threadIdx.x;  // over B*T*P
    const int p   = (int)(idx & (Pc - 1));
    const int rid = (int)(idx >> 8);        // b*T + t   (P = 256)
    const int t   = rid & (Tc - 1);
    const int b   = rid >> 12;              // T = 4096
    const int ch  = t / CHUNK;

    const float2 h  = *(const float2*)(hin + ((size_t)(b * NCH + ch) * Pc + p) * 2);
    const float  cA = angles[idx];
    const float  cD = cumD[rid];
    float s, c;
    __sincosf(cA, &s, &c);

    float2* op = (float2*)(out + (size_t)rid * Dc + 2 * p);
    float2  v  = *op;
    v.x += cD * (c * h.x - s * h.y);
    v.y += cD * (s * h.x + c * h.y);
    *op = v;
}

// ---------------------------------------------------------------------------
extern "C" void kernel_launch(void* const* d_in, const int* in_sizes, int n_in,
                              void* d_out, int out_size, void* d_ws, size_t ws_size,
                              hipStream_t stream) {
    (void)in_sizes; (void)n_in; (void)out_size; (void)ws_size;

    const float* x    = (const float*)d_in[0];
    const float* Wang = (const float*)d_in[1];
    const float* bang = (const float*)d_in[2];
    const float* Wdec = (const float*)d_in[3];
    const float* bdec = (const float*)d_in[4];
    const float* Winp = (const float*)d_in[5];
    const float* binp = (const float*)d_in[6];
    float* out = (float*)d_out;

    // Workspace layout (floats): angles | decays | cumD | sumA | hin | packW (~34.5 MB)
    float* ws     = (float*)d_ws;
    float* angles = ws;                                    // BT * P
    float* decays = angles + (size_t)BT * Pc;              // BT
    float* cumD   = decays + BT;                           // BT
    float* sumA   = cumD   + BT;                           // B*NCH*P
    float* hin    = sumA   + (size_t)Bc * NCH * Pc;        // B*NCH*P*2
    float* packW  = hin    + (size_t)Bc * NCH * Pc * 2;    // 48*64*32*4

    packW_kernel<<<dim3((NTILES * KQ2 * 32) / 256), dim3(256), 0, stream>>>(
        Wang, Winp, packW);
    proj_wmma_kernel<<<dim3(BT / 16), dim3(256), 0, stream>>>(
        x, packW, bang, binp, angles, out);
    decay_kernel<<<dim3(BT / 256), dim3(256), 0, stream>>>(x, Wdec, bdec, decays);
    scan_local_kernel<<<dim3(Bc * NCH), dim3(256), 0, stream>>>(
        angles, decays, cumD, sumA, out);
    scan_carry_kernel<<<dim3(Bc), dim3(256), 0, stream>>>(sumA, cumD, out, hin);
    fixup_kernel<<<dim3((Bc * Tc * Pc) / 256), dim3(256), 0, stream>>>(
        angles, cumD, hin, out);
}